// QuantumAttentionLayer_65481071397460
// MI455X (gfx1250) — compile-verified
//
#include <hip/hip_runtime.h>

// cumprod(cos(x), axis=-1) over (16, 8192, 512) f32.
// Memory-bound streamer: async global->LDS double-buffered prefetch (CDNA5
// ASYNCcnt path), native v_cos_f32, per-lane 16-elem serial cumprod,
// raw-bpermute wave32 scan, nontemporal b128 stores.

typedef __attribute__((ext_vector_type(4))) float f32x4;

#define NROWS (16 * 8192)   // 131072 rows
#define ROW   512           // floats per row (2048 bytes)
#define WPB   8             // waves per block
#define BLOCK 256           // threads per block
#define NBLK  2048          // 16384 waves total -> exactly 8 rows per wave

// One row = 2KB = 4 x (32 lanes x 16B) async b128 transfers.
// GVS mode: addr = SADDR(64b, block-uniform) + VADDR(32b, per-lane) + IOFFSET.
// IOFFSET is added to BOTH the global and the LDS address, so one VADDR/VDST
// setup covers the whole row with 4 instructions. NT hint: touch-once stream.
#define ASYNC_ROW(ldsoff, srcptr)                                                            \
    asm volatile("global_load_async_to_lds_b128 %0, %1, %2 offset:0 th:TH_LOAD_NT\n\t"       \
                 "global_load_async_to_lds_b128 %0, %1, %2 offset:512 th:TH_LOAD_NT\n\t"     \
                 "global_load_async_to_lds_b128 %0, %1, %2 offset:1024 th:TH_LOAD_NT\n\t"    \
                 "global_load_async_to_lds_b128 %0, %1, %2 offset:1536 th:TH_LOAD_NT"        \
                 :: "v"(ldsoff), "v"(vga), "s"(srcptr)                                       \
                 : "memory")

// Native cos: inputs are N(0,1) (|x| <~ 6), well inside v_cos_f32's domain.
__device__ __forceinline__ float fast_cos(float x) {
    return __builtin_amdgcn_cosf(x * 0.15915494309189535f); // x / (2*pi)
}

__device__ __forceinline__ float lane_shift(float v, int idx) {
    return __int_as_float(__builtin_amdgcn_ds_bpermute(idx, __float_as_int(v)));
}

__global__ __launch_bounds__(BLOCK) void cumprod_cos_kernel(
    const float* __restrict__ x, float* __restrict__ y)
{
    // [wave][ping-pong buf][row]: 8 * 2 * 512 floats = 32 KB
    __shared__ float lds[WPB * 2 * ROW];

    const unsigned lane = threadIdx.x & 31u;
    const unsigned wid  = threadIdx.x >> 5;

    // Per-lane 32-bit offset into the block's 8-row slab (wave wid -> row wid).
    const unsigned vga = wid * (ROW * 4u) + lane * 16u;

    // LDS byte addresses of this wave's two row buffers (per-lane 16B chunk).
    const unsigned lbase0 =
        (unsigned)(size_t)(&lds[wid * 2 * ROW]) + lane * 16u;
    const unsigned lbase1 = lbase0 + (unsigned)(ROW * 4);

    const unsigned rows_per_iter = (unsigned)gridDim.x * WPB;
    const unsigned brow0 = blockIdx.x * WPB;   // block's first row this sweep

    // Loop-invariant scan helpers: bpermute byte-indices (wrap via bits[6:2],
    // out-of-range lanes are masked by the guards) + hoisted lane guards.
    const int i1  = ((int)lane - 1)  << 2;
    const int i2  = ((int)lane - 2)  << 2;
    const int i4  = ((int)lane - 4)  << 2;
    const int i8  = ((int)lane - 8)  << 2;
    const int i16 = ((int)lane - 16) << 2;
    const bool g1  = lane >= 1,  g2 = lane >= 2, g4 = lane >= 4;
    const bool g8  = lane >= 8,  g16 = lane >= 16;

    // Prologue: prefetch first row slab for this wave.
    if (brow0 < NROWS) {
        const float* s0 = x + (size_t)brow0 * ROW;
        ASYNC_ROW(lbase0, s0);
    }

    unsigned cur = 0;
    for (unsigned brow = brow0; brow < NROWS; brow += rows_per_iter) {
        const unsigned nbrow = brow + rows_per_iter;
        if (nbrow < NROWS) {
            // Prefetch next sweep's row into the other buffer, then wait for
            // the current row: async loads complete in order, so ASYNCcnt<=4
            // means the first 4 (current row) have landed in LDS.
            const float* s1 = x + (size_t)nbrow * ROW;
            const unsigned dst = cur ? lbase0 : lbase1;
            ASYNC_ROW(dst, s1);
            asm volatile("s_wait_asynccnt 4" ::: "memory");
        } else {
            asm volatile("s_wait_asynccnt 0" ::: "memory");
        }

        // ---- read this wave's row (16 floats/lane) from LDS: 4x ds_load_b128
        const f32x4* lr = (const f32x4*)(lds + wid * 2 * ROW + cur * ROW) + lane * 4;
        f32x4 a0 = lr[0];
        f32x4 a1 = lr[1];
        f32x4 a2 = lr[2];
        f32x4 a3 = lr[3];

        float v[16] = { a0.x, a0.y, a0.z, a0.w,
                        a1.x, a1.y, a1.z, a1.w,
                        a2.x, a2.y, a2.z, a2.w,
                        a3.x, a3.y, a3.z, a3.w };

        // ---- local inclusive cumprod of cos over this lane's 16 elements
        float c[16];
        float r = 1.0f;
        #pragma unroll
        for (int k = 0; k < 16; ++k) {
            r *= fast_cos(v[k]);
            c[k] = r;
        }

        // ---- wave32 inclusive scan of per-lane totals (Hillis-Steele),
        // raw ds_bpermute with precomputed indices + hoisted guards:
        // 1 bpermute + 1 cndmask + 1 mul per step.
        float p = r, t;
        t = lane_shift(p, i1);  p *= g1  ? t : 1.0f;
        t = lane_shift(p, i2);  p *= g2  ? t : 1.0f;
        t = lane_shift(p, i4);  p *= g4  ? t : 1.0f;
        t = lane_shift(p, i8);  p *= g8  ? t : 1.0f;
        t = lane_shift(p, i16); p *= g16 ? t : 1.0f;
        // exclusive prefix for this lane (lane 0 -> 1.0)
        float e = lane_shift(p, i1);
        e = g1 ? e : 1.0f;

        // ---- scale + nontemporal b128 stores (write-once stream)
        const unsigned row = brow + wid;
        float* op = y + (size_t)row * ROW + lane * 16u;
        #pragma unroll
        for (int q = 0; q < 4; ++q) {
            f32x4 o = { c[4 * q + 0] * e, c[4 * q + 1] * e,
                        c[4 * q + 2] * e, c[4 * q + 3] * e };
            __builtin_nontemporal_store(o, (f32x4*)(op + 4 * q));
        }

        cur ^= 1u;
    }
}

extern "C" void kernel_launch(void* const* d_in, const int* in_sizes, int n_in,
                              void* d_out, int out_size, void* d_ws, size_t ws_size,
                              hipStream_t stream) {
    const float* x = (const float*)d_in[0];
    float*       y = (float*)d_out;
    cumprod_cos_kernel<<<NBLK, BLOCK, 0, stream>>>(x, y);
}